// Caps2dMatwo_69784628625918
// MI455X (gfx1250) — compile-verified
//
#include <hip/hip_runtime.h>
#include <math.h>
#include <stdint.h>

// CDNA5 / gfx1250, wave32. Fully fused Caps2dMatwo:
//   conv(5x5,1->8) via V_WMMA_F32_16X16X4_F32 chained over K,
//   4x4 capsule matmuls + 3-round sigmoid routing in VALU f32,
//   u_hat kept entirely in LDS (64KB tile) -- avoids 151MB HBM intermediate.
// Ideal traffic ~57MB @ 23.3TB/s; compute ~5 GFLOP, conv goes to matrix pipe.

typedef __attribute__((ext_vector_type(2))) float v2f;
typedef __attribute__((ext_vector_type(8))) float v8f;

#define HW 96
#define PLANE (HW*HW)          // 9216
#define ROUTS 3

__device__ __forceinline__ float sigmoidf_(float x) {
    return 1.0f / (1.0f + expf(-x));
}

__global__ __launch_bounds__(32)
void caps_fused_kernel(const float* __restrict__ x,    // (4,4,32,96,96)
                       const float* __restrict__ Wc,   // (4,5,5,1,8)
                       const float* __restrict__ Wp,   // (4,16,8)
                       const float* __restrict__ Wa,   // (4,16,8)
                       const float* __restrict__ Ba,   // (4,8)
                       float* __restrict__ out)        // (4,8,32,96,96)
{
    // ---------------- LDS ----------------
    __shared__ float lds_w[800];          // W_conv: (i*25+k)*8+t1
    __shared__ float lds_mpos[512];       // normalized m_pos: ((i*8+g)*4+k)*4+n
    __shared__ float lds_mapp[512];       // m_app
    __shared__ float lds_bapp[32];        // b_app[i*8+g]
    __shared__ float lds_in[32 * 64];     // input halo [z][y*8+x], 8x8 per z
    __shared__ float lds_us[32 * 128];    // conv out   [z][p*8+t1]
    __shared__ float lds_uhat[16*32*32];  // u_hat      [p][zc][o*8+t]  (64KB)

    const int lane = threadIdx.x;
    const int n    = blockIdx.z;
    const int h0   = blockIdx.y * 4;
    const int w0   = blockIdx.x * 4;

    // ---------------- small-weight preprocessing ----------------
    for (int j = lane; j < 800; j += 32) lds_w[j] = Wc[j];
    {
        // one (i,g) pair per lane: i=lane>>3, g=lane&7
        const int i = lane >> 3, g = lane & 7;
        float m[16];
        #pragma unroll
        for (int j = 0; j < 16; j++) m[j] = Wp[i*128 + g*16 + j];
        #pragma unroll
        for (int c = 0; c < 4; c++) {
            float s = m[0*4+c]*m[0*4+c] + m[1*4+c]*m[1*4+c]
                    + m[2*4+c]*m[2*4+c] + m[3*4+c]*m[3*4+c];
            s = fmaxf(s, 1e-12f);
            float inv = 1.0f / sqrtf(s);
            #pragma unroll
            for (int r = 0; r < 4; r++)
                lds_mpos[lane*16 + r*4 + c] = m[r*4+c] * inv;
        }
        #pragma unroll
        for (int j = 0; j < 16; j++) lds_mapp[lane*16 + j] = Wa[i*128 + g*16 + j];
        lds_bapp[lane] = Ba[lane];
    }
    __syncthreads();

    // ---------------- per-lane WMMA constants ----------------
    // A layout (16x4 f32): lanes 0-15 -> M=lane, K={0,1}; lanes 16-31 -> M=lane-16, K={2,3}
    const int mrow  = lane & 15;          // pixel row of A / C column id source
    const int py    = mrow >> 2, px = mrow & 3;
    const int khalf = (lane < 16) ? 0 : 2;
    const int ncol  = lane & 15;          // B/C column

    int aoff[14];                         // LDS halo offsets for the 2 taps per K-step
    #pragma unroll
    for (int kk = 0; kk < 7; kk++) {
        #pragma unroll
        for (int j = 0; j < 2; j++) {
            int k = kk*4 + khalf + j;
            if (k < 25) { int dy = k / 5, dx = k % 5; aoff[kk*2+j] = (py+dy)*8 + (px+dx); }
            else        { aoff[kk*2+j] = -1; }
        }
    }

    // ================ loop over input capsule-type i (t0) ================
    #pragma unroll 1
    for (int i = 0; i < 4; i++) {
        // ---- stage 32z x 8x8 halo into LDS (async global->LDS; zero SAME padding) ----
        const float* base = x + ((size_t)(n*4 + i)) * 32 * PLANE;
        #pragma unroll 1
        for (int j = 0; j < 64; j++) {
            int idx = lane + j * 32;               // 0..2047
            int z = idx >> 6, rem = idx & 63, yy = rem >> 3, xx = rem & 7;
            int hh = h0 - 2 + yy, ww = w0 - 2 + xx;
            if (hh >= 0 && hh < HW && ww >= 0 && ww < HW) {
                const float* g = base + z*PLANE + hh*HW + ww;
                unsigned int loff = (unsigned int)(uintptr_t)(&lds_in[idx]);
                unsigned long long ga = (unsigned long long)(uintptr_t)g;
                asm volatile("global_load_async_to_lds_b32 %0, %1, off"
                             :: "v"(loff), "v"(ga) : "memory");
            } else {
                lds_in[idx] = 0.0f;
            }
        }
        asm volatile("s_wait_asynccnt 0" ::: "memory");
        __syncthreads();

        // ---- per-lane B (4x16 f32) for the 7 chained K-steps ----
        v2f Bk[7];
        #pragma unroll
        for (int kk = 0; kk < 7; kk++) {
            float bx = 0.0f, by = 0.0f;
            int k0 = kk*4 + khalf;
            if (ncol < 8) {
                if (k0     < 25) bx = lds_w[(i*25 + k0    )*8 + ncol];
                if (k0 + 1 < 25) by = lds_w[(i*25 + k0 + 1)*8 + ncol];
            }
            Bk[kk].x = bx; Bk[kk].y = by;
        }

        // ---- conv: 7x V_WMMA_F32_16X16X4_F32 per z, C = 16 pixels x 8 channels ----
        #pragma unroll 1
        for (int z = 0; z < 32; z++) {
            v8f c = {0.f,0.f,0.f,0.f,0.f,0.f,0.f,0.f};
            const float* zi = &lds_in[z * 64];
            #pragma unroll
            for (int kk = 0; kk < 7; kk++) {
                v2f a;
                a.x = (aoff[kk*2+0] >= 0) ? zi[aoff[kk*2+0]] : 0.0f;
                a.y = (aoff[kk*2+1] >= 0) ? zi[aoff[kk*2+1]] : 0.0f;
                c = __builtin_amdgcn_wmma_f32_16x16x4_f32(
                        /*neg_a=*/false, a, /*neg_b=*/false, Bk[kk],
                        /*c_mod=*/(short)0, c, /*reuse_a=*/false, /*reuse_b=*/false);
            }
            // C layout: lanes0-15 col=lane rows r=0..7; lanes16-31 rows 8..15
            if (ncol < 8) {
                int rbase = (lane < 16) ? 0 : 8;
                #pragma unroll
                for (int r = 0; r < 8; r++)
                    lds_us[z*128 + (rbase + r)*8 + ncol] = c[r];
            }
        }
        __syncthreads();

        // ---- 4x4 capsule matmuls: 256 (pixel,group) tasks over 32 lanes ----
        #pragma unroll 1
        for (int it = 0; it < 8; it++) {
            int task = lane + it*32;
            int p  = task >> 4;
            int g2 = task & 15;
            int g  = g2 & 7;
            bool isApp = (g2 >= 8);
            int t1 = (isApp ? 4 : 0) + (g >> 1);
            int zb = (g & 1) * 16;
            float badd = isApp ? lds_bapp[i*8 + g] : 0.0f;

            float u[16];
            #pragma unroll
            for (int r = 0; r < 4; r++)
                #pragma unroll
                for (int k = 0; k < 4; k++)
                    u[r*4+k] = lds_us[(zb + r*4 + k)*128 + p*8 + t1] + badd;

            const float* M = (isApp ? lds_mapp : lds_mpos) + (i*8 + g)*16;
            int ppy = p >> 2, ppx = p & 3;
            float cx = (float)(w0 + ppx) / 96.0f;   // coord add (post-normalization)
            float cy = (float)(h0 + ppy) / 96.0f;

            float* uh = &lds_uhat[(p*32 + (isApp ? 16 : 0))*32 + i*8 + g];
            #pragma unroll
            for (int r = 0; r < 4; r++) {
                #pragma unroll
                for (int nn = 0; nn < 4; nn++) {
                    float s = 0.0f;
                    #pragma unroll
                    for (int k = 0; k < 4; k++) s += u[r*4+k] * M[k*4+nn];
                    if (!isApp) {
                        if (nn == 0) s += u[r*4+3] * cx;
                        if (nn == 1) s += u[r*4+3] * cy;
                    }
                    uh[(r*4 + nn)*32] = s;
                }
            }
        }
        __syncthreads();
    } // i

    // ================ dynamic routing: lane = output capsule zc ================
    const int  zc    = lane;
    const bool isPos = (lane < 16);      // zc<16: psquash half, zc>=16: matwo half
    #pragma unroll 1
    for (int p = 0; p < 16; p++) {
        float uu[32];
        #pragma unroll
        for (int j = 0; j < 32; j++) uu[j] = lds_uhat[(p*32 + zc)*32 + j];
        float bb[32];
        #pragma unroll
        for (int j = 0; j < 32; j++) bb[j] = 0.0f;

        float vv[8];
        #pragma unroll 1
        for (int iter = 0; iter < ROUTS; iter++) {
            float pt[8];
            #pragma unroll
            for (int t = 0; t < 8; t++) {
                float s = 0.0f;
                #pragma unroll
                for (int o = 0; o < 4; o++) s += uu[o*8+t] * sigmoidf_(bb[o*8+t]);
                pt[t] = s;
            }
            // per-t reduction over the 16 lanes of each half:
            //   pos: max|p|   app: sum p^2
            float red[8];
            #pragma unroll
            for (int t = 0; t < 8; t++) red[t] = isPos ? fabsf(pt[t]) : pt[t]*pt[t];
            #pragma unroll
            for (int m = 1; m < 16; m <<= 1) {
                #pragma unroll
                for (int t = 0; t < 8; t++) {
                    float o = __shfl_xor(red[t], m, 32);
                    red[t] = isPos ? fmaxf(red[t], o) : (red[t] + o);
                }
            }
            #pragma unroll
            for (int t = 0; t < 8; t++) {
                if (isPos) {
                    vv[t] = pt[t] / red[t];                    // psquash
                } else {
                    float n2 = red[t];
                    float nn = sqrtf(n2 + 1e-9f);              // matwo squash
                    vv[t] = n2 / (1.0f + n2) * pt[t] / nn;
                }
            }
            if (iter < ROUTS - 1) {
                // rout[o,t] = (sum_{z<16} u*v) * (sum_{z>=16} u*v)
                #pragma unroll
                for (int j = 0; j < 32; j++) {
                    float s = uu[j] * vv[j & 7];
                    #pragma unroll
                    for (int m = 1; m < 16; m <<= 1) s += __shfl_xor(s, m, 32);
                    float other = __shfl_xor(s, 16, 32);
                    bb[j] += s * other;
                }
            }
        }

        // out[n, t, zc, h, w]
        int hh = h0 + (p >> 2), ww = w0 + (p & 3);
        #pragma unroll
        for (int t = 0; t < 8; t++) {
            out[(((size_t)(n*8 + t))*32 + zc) * PLANE + hh*HW + ww] = vv[t];
        }
    }
}

extern "C" void kernel_launch(void* const* d_in, const int* in_sizes, int n_in,
                              void* d_out, int out_size, void* d_ws, size_t ws_size,
                              hipStream_t stream) {
    (void)in_sizes; (void)n_in; (void)out_size; (void)d_ws; (void)ws_size;
    const float* x  = (const float*)d_in[0];
    const float* Wc = (const float*)d_in[1];
    const float* Wp = (const float*)d_in[2];
    const float* Wa = (const float*)d_in[3];
    const float* Ba = (const float*)d_in[4];
    float* out = (float*)d_out;
    dim3 grid(96/4, 96/4, 4);   // (w tiles, h tiles, N)
    dim3 block(32);             // one wave32 per workgroup
    hipLaunchKernelGGL(caps_fused_kernel, grid, block, 0, stream,
                       x, Wc, Wp, Wa, Ba, out);
}